// PositionEncoder_10411000726289
// MI455X (gfx1250) — compile-verified
//
#include <hip/hip_runtime.h>
#include <hip/hip_bf16.h>
#include <math.h>

#define Kn   20
#define Bc   4
#define Mc   2048
#define Nc   8192
#define Rtot (Bc*Mc)          // 8192 rows (b*M+m)
#define C8   32
#define C4   64
#define COUT 256

typedef float v2f __attribute__((ext_vector_type(2)));
typedef float v8f __attribute__((ext_vector_type(8)));

__device__ __forceinline__ float gelu_exact(float x) {
    return 0.5f * x * (1.0f + erff(x * 0.70710678118654752440f));
}

// ---------------------------------------------------------------- zero stats
__global__ void zero_kernel(float* __restrict__ p, int n) {
    int t = blockIdx.x * blockDim.x + threadIdx.x;
    if (t < n) p[t] = 0.0f;
}

// --------------------------------------------- wave-per-row ballot select +
// 9->32 layer0 (each lane owns one output channel; K neighbors via LDS idx)
__global__ void select_layer0(const float* __restrict__ dist,
                              const float* __restrict__ radius,
                              const float* __restrict__ xyz,
                              const float* __restrict__ centroid,
                              const float* __restrict__ w_xyz,
                              const float* __restrict__ b_xyz,
                              float* __restrict__ h0) {
    __shared__ int s_idx[8][Kn];
    const int wave  = threadIdx.x >> 5;
    const int lane  = threadIdx.x & 31;
    const int rowid = blockIdx.x * 8 + wave;        // b*M + m, grid sized exactly
    const int b     = rowid / Mc;

    const float* drow = dist + (size_t)rowid * Nc;
    const float  r    = radius[rowid];

    int have = 0;
    for (int base = 0; base < Nc; base += 32) {
        const int  n   = base + lane;
        const bool hit = (drow[n] <= r);
        const unsigned mm = (unsigned)__ballot(hit);
        if (hit) {
            int slot = have + __popc(mm & ((1u << lane) - 1u));
            if (slot < Kn) s_idx[wave][slot] = n;
        }
        have += __popc(mm);
        if (have >= Kn) break;                       // early exit: dist traffic ~5 MB not 256 MB
    }
    __syncthreads();
    if (lane == 0) {
        int fill = (have > 0) ? s_idx[wave][0] : (Nc - 1);   // JAX clipped-gather semantics
        for (int k = have; k < Kn; ++k) s_idx[wave][k] = fill;
    }
    __syncthreads();

    // lane == output channel c (C8 == 32 == wave width)
    float wrow[9];
    #pragma unroll
    for (int j = 0; j < 9; ++j) wrow[j] = w_xyz[lane * 9 + j];
    const float bias = b_xyz[lane];

    const float cx = centroid[rowid * 3 + 0];
    const float cy = centroid[rowid * 3 + 1];
    const float cz = centroid[rowid * 3 + 2];
    const float* xb = xyz + (size_t)b * Nc * 3;

    for (int k = 0; k < Kn; ++k) {
        const int n = s_idx[wave][k];
        const float px = xb[n * 3 + 0];
        const float py = xb[n * 3 + 1];
        const float pz = xb[n * 3 + 2];
        float h = bias;
        h = fmaf(wrow[0], cx, h);  h = fmaf(wrow[1], cy, h);  h = fmaf(wrow[2], cz, h);
        h = fmaf(wrow[3], px, h);  h = fmaf(wrow[4], py, h);  h = fmaf(wrow[5], pz, h);
        h = fmaf(wrow[6], px - cx, h); h = fmaf(wrow[7], py - cy, h); h = fmaf(wrow[8], pz - cz, h);
        h0[((size_t)rowid * Kn + k) * C8 + lane] = h;
    }
}

// ------------------------------------------------ per-channel sum / sumsq
// requires Cc in {32,64,256}, blockDim.x == 256
__global__ void colreduce(const float* __restrict__ src, int rows, int Cc,
                          float* __restrict__ osum, float* __restrict__ osq) {
    const int c   = threadIdx.x % Cc;
    const int sub = threadIdx.x / Cc;
    const int rpb = blockDim.x / Cc;
    float s = 0.0f, sq = 0.0f;
    for (long long row = (long long)blockIdx.x * rpb + sub; row < rows;
         row += (long long)gridDim.x * rpb) {
        const float v = src[row * Cc + c];
        s += v; sq += v * v;
    }
    __shared__ float red[2][256];
    red[0][threadIdx.x] = s; red[1][threadIdx.x] = sq;
    __syncthreads();
    if (threadIdx.x < Cc) {
        for (int i = 1; i < rpb; ++i) {
            s  += red[0][i * Cc + threadIdx.x];
            sq += red[1][i * Cc + threadIdx.x];
        }
        atomicAdd(&osum[c], s);
        atomicAdd(&osq[c], sq);
    }
}

// ----------------------------------------- fold BN into per-channel affine
__global__ void finalize_bn(const float* __restrict__ sum, const float* __restrict__ sq,
                            float cntInv, const float* __restrict__ gamma,
                            const float* __restrict__ beta,
                            float* __restrict__ scale, float* __restrict__ shift, int Cc) {
    const int c = blockIdx.x * blockDim.x + threadIdx.x;
    if (c < Cc) {
        const float mean = sum[c] * cntInv;
        const float var  = sq[c] * cntInv - mean * mean;
        const float inv  = rsqrtf(var + 1e-5f);
        const float sc   = gamma[c] * inv;
        scale[c] = sc;
        shift[c] = beta[c] - mean * sc;
    }
}

// -------------------------------------------- BN0 + GELU + max over K
__global__ void pool_kernel(const float* __restrict__ h0, const float* __restrict__ scale0,
                            const float* __restrict__ shift0, float* __restrict__ pooled) {
    const int t = blockIdx.x * blockDim.x + threadIdx.x;   // Rtot*32 threads exactly
    const int c = t & 31;
    const int row = t >> 5;
    const float sc = scale0[c], sh = shift0[c];
    const float* p = h0 + (size_t)row * Kn * C8 + c;
    float m = -INFINITY;
    #pragma unroll
    for (int k = 0; k < Kn; ++k) {
        m = fmaxf(m, gelu_exact(fmaf(p[k * C8], sc, sh)));
    }
    pooled[t] = m;
}

// -------------------------------------------- GEMM1: (8192x32) @ (32x64)^T
// wave tile 16x16, K stepped by 4 via V_WMMA_F32_16X16X4_F32 (pure f32)
__global__ void gemm1_wmma(const float* __restrict__ pooled, const float* __restrict__ w1,
                           float* __restrict__ h1) {
    const int wave = blockIdx.x * (blockDim.x >> 5) + (threadIdx.x >> 5); // 2048 waves
    const int lane = threadIdx.x & 31;
    const int rowtile = wave & 511;        // 512 row tiles
    const int coltile = wave >> 9;         // 4 col tiles
    const int row0 = rowtile * 16, n0 = coltile * 16;
    const int lhalf = lane & 15;
    const int khi   = (lane >> 4) * 2;     // lanes 16..31 hold K+2,K+3

    const float* arow = pooled + (size_t)(row0 + lhalf) * C8;
    const float* brow = w1     + (size_t)(n0   + lhalf) * C8;

    v8f c = {0.f,0.f,0.f,0.f,0.f,0.f,0.f,0.f};
    #pragma unroll
    for (int k0 = 0; k0 < C8; k0 += 4) {
        const int ka = k0 + khi;
        v2f a; a.x = arow[ka]; a.y = arow[ka + 1];
        v2f b; b.x = brow[ka]; b.y = brow[ka + 1];
        c = __builtin_amdgcn_wmma_f32_16x16x4_f32(false, a, false, b,
                                                  (short)0, c, false, false);
    }
    const int mbase = row0 + (lane >> 4) * 8;
    const int n     = n0 + lhalf;
    #pragma unroll
    for (int r2 = 0; r2 < 8; ++r2)
        h1[(size_t)(mbase + r2) * C4 + n] = c[r2];
}

// ------------------------- GEMM2: act1 = gelu(bn1(h1)) fused into A loads,
// (8192x64) @ (64x256)^T, V_WMMA_F32_16X16X4_F32
__global__ void gemm2_wmma(const float* __restrict__ h1, const float* __restrict__ w2,
                           const float* __restrict__ scale1, const float* __restrict__ shift1,
                           float* __restrict__ h2) {
    const int wave = blockIdx.x * (blockDim.x >> 5) + (threadIdx.x >> 5); // 8192 waves
    const int lane = threadIdx.x & 31;
    const int rowtile = wave & 511;        // 512 row tiles
    const int coltile = wave >> 9;         // 16 col tiles
    const int row0 = rowtile * 16, n0 = coltile * 16;
    const int lhalf = lane & 15;
    const int khi   = (lane >> 4) * 2;

    const float* arow = h1 + (size_t)(row0 + lhalf) * C4;
    const float* brow = w2 + (size_t)(n0   + lhalf) * C4;

    v8f c = {0.f,0.f,0.f,0.f,0.f,0.f,0.f,0.f};
    #pragma unroll
    for (int k0 = 0; k0 < C4; k0 += 4) {
        const int ka = k0 + khi;
        const float x0 = fmaf(arow[ka],     scale1[ka],     shift1[ka]);
        const float x1 = fmaf(arow[ka + 1], scale1[ka + 1], shift1[ka + 1]);
        v2f a; a.x = gelu_exact(x0); a.y = gelu_exact(x1);
        v2f b; b.x = brow[ka]; b.y = brow[ka + 1];
        c = __builtin_amdgcn_wmma_f32_16x16x4_f32(false, a, false, b,
                                                  (short)0, c, false, false);
    }
    const int mbase = row0 + (lane >> 4) * 8;
    const int n     = n0 + lhalf;
    #pragma unroll
    for (int r2 = 0; r2 < 8; ++r2)
        h2[(size_t)(mbase + r2) * COUT + n] = c[r2];
}

// -------------------------------------------- final BN2 affine -> d_out
__global__ void bn_apply(const float* __restrict__ h2, const float* __restrict__ scale2,
                         const float* __restrict__ shift2, float* __restrict__ out, int total) {
    const int t = blockIdx.x * blockDim.x + threadIdx.x;
    if (t < total) {
        const int c = t & (COUT - 1);
        out[t] = fmaf(h2[t], scale2[c], shift2[c]);
    }
}

// ---------------------------------------------------------------- launcher
extern "C" void kernel_launch(void* const* d_in, const int* in_sizes, int n_in,
                              void* d_out, int out_size, void* d_ws, size_t ws_size,
                              hipStream_t stream) {
    const float* centroid = (const float*)d_in[0];
    const float* xyz      = (const float*)d_in[1];
    const float* radius   = (const float*)d_in[2];
    const float* dist     = (const float*)d_in[3];
    const float* w_xyz    = (const float*)d_in[4];
    const float* b_xyz    = (const float*)d_in[5];
    const float* g0       = (const float*)d_in[6];
    const float* be0      = (const float*)d_in[7];
    const float* w1       = (const float*)d_in[8];
    const float* g1       = (const float*)d_in[9];
    const float* be1      = (const float*)d_in[10];
    const float* w2       = (const float*)d_in[11];
    const float* g2       = (const float*)d_in[12];
    const float* be2      = (const float*)d_in[13];
    float* out = (float*)d_out;
    float* ws  = (float*)d_ws;

    // workspace layout (floats)
    float* h0     = ws;                                   // 8192*20*32 = 5,242,880
    float* pooled = h0 + (size_t)Rtot * Kn * C8;          // 8192*32
    float* h1     = pooled + (size_t)Rtot * C8;           // 8192*64
    float* h2     = h1 + (size_t)Rtot * C4;               // 8192*256
    float* stats  = h2 + (size_t)Rtot * COUT;
    float* sum0 = stats;        float* sq0 = sum0 + C8;
    float* sum1 = sq0 + C8;     float* sq1 = sum1 + C4;
    float* sum2 = sq1 + C4;     float* sq2 = sum2 + COUT;
    float* scale0 = sq2 + COUT; float* shift0 = scale0 + C8;
    float* scale1 = shift0 + C8;  float* shift1 = scale1 + C4;
    float* scale2 = shift1 + C4;  float* shift2 = scale2 + COUT;
    const int nstats = 2 * (C8 + C4 + COUT);              // 704

    zero_kernel<<<(nstats + 255) / 256, 256, 0, stream>>>(stats, nstats);

    // 1 wave per (b,m) row: ballot scan with early exit + layer0
    select_layer0<<<Rtot / 8, 256, 0, stream>>>(dist, radius, xyz, centroid,
                                                w_xyz, b_xyz, h0);

    colreduce<<<256, 256, 0, stream>>>(h0, Rtot * Kn, C8, sum0, sq0);
    finalize_bn<<<1, 256, 0, stream>>>(sum0, sq0, 1.0f / (Rtot * Kn), g0, be0,
                                       scale0, shift0, C8);

    pool_kernel<<<(Rtot * C8) / 256, 256, 0, stream>>>(h0, scale0, shift0, pooled);

    gemm1_wmma<<<(512 * 4) / 8, 256, 0, stream>>>(pooled, w1, h1);       // 256 blocks
    colreduce<<<64, 256, 0, stream>>>(h1, Rtot, C4, sum1, sq1);
    finalize_bn<<<1, 256, 0, stream>>>(sum1, sq1, 1.0f / Rtot, g1, be1,
                                       scale1, shift1, C4);

    gemm2_wmma<<<(512 * 16) / 8, 256, 0, stream>>>(h1, w2, scale1, shift1, h2); // 1024 blocks
    colreduce<<<64, 256, 0, stream>>>(h2, Rtot, COUT, sum2, sq2);
    finalize_bn<<<1, 256, 0, stream>>>(sum2, sq2, 1.0f / Rtot, g2, be2,
                                       scale2, shift2, COUT);

    bn_apply<<<(Rtot * COUT) / 256, 256, 0, stream>>>(h2, scale2, shift2, out,
                                                      Rtot * COUT);
}